// Embedding_68324339745060
// MI455X (gfx1250) — compile-verified
//
#include <hip/hip_runtime.h>

// Embedding gather + unstable softmax (exp / sum(exp)) over last axis.
// Outputs: logits [R,256] then probs [R,256], concatenated in d_out.
//
// MI455X reasoning: traffic = 256 MiB NT stores + ~128 MiB gathered reads.
// Embedding table (131 MiB) fits in the 192 MB L2, so outputs use
// non-temporal stores to avoid evicting it; gathered rows are prefetched
// with CDNA5 async global->LDS (ASYNCcnt) into a per-wave double buffer,
// and the scalar index load is pipelined one further iteration ahead so
// neither SMEM nor HBM latency sits on the critical path.

typedef __attribute__((ext_vector_type(4))) float v4f;

#define EMBED 256
#define WAVES_PER_BLOCK 8
#define BLOCKS 2048

// One xor-butterfly reduction step: 1 ds_swizzle + 1 v_add (all lanes active).
#define XOR_ADD(s, pat) \
    (s) += __int_as_float(__builtin_amdgcn_ds_swizzle(__float_as_int(s), (pat)))

__global__ __launch_bounds__(256, 4) void embed_softmax_kernel(
    const int* __restrict__ idx,
    const float* __restrict__ emb,
    float* __restrict__ logits,
    float* __restrict__ probs,
    int R)
{
    // 8 waves * 2 buffers * 256 floats = 16 KiB LDS per block
    __shared__ __align__(16) float smem[WAVES_PER_BLOCK * 2 * EMBED];

    const int lane = threadIdx.x & 31;
    const int wib  = __builtin_amdgcn_readfirstlane((int)(threadIdx.x >> 5));
    float* wbuf = smem + wib * (2 * EMBED);

    const int wavesTotal = gridDim.x * WAVES_PER_BLOCK;
    const int w = blockIdx.x * WAVES_PER_BLOCK + wib;

    // Async prefetch of embedding row `v` into LDS buffer `bufsel`.
    // Each lane moves 16B; offset:512 covers the second half of the row
    // (INST_OFFSET is added to BOTH the LDS and the global address).
    auto prefetch_row = [&](int v, int bufsel) {
        unsigned long long ga = (unsigned long long)(const void*)
            (emb + (size_t)(unsigned)v * EMBED + lane * 4);
        unsigned ldsa = (unsigned)(size_t)(wbuf + bufsel * EMBED + lane * 4);
        asm volatile(
            "global_load_async_to_lds_b128 %0, %1, off\n\t"
            "global_load_async_to_lds_b128 %0, %1, off offset:512"
            :: "v"(ldsa), "v"(ga)
            : "memory");
    };

    int pb = 0;
    int vnext = 0;
    if (w < R) {
        prefetch_row(idx[w], 0);               // row 0 in flight
        if (w + wavesTotal < R)
            vnext = idx[w + wavesTotal];       // index for row 1 in flight
    }

    for (int r = w; r < R; r += wavesTotal) {
        const int rn = r + wavesTotal;
        if (rn < R) {
            prefetch_row(vnext, pb ^ 1);       // launch next row's gather
            const int rn2 = rn + wavesTotal;
            if (rn2 < R) vnext = idx[rn2];     // scalar load, consumed next iter
            // 4 async loads outstanding; current row's 2 complete in order.
            asm volatile("s_wait_asynccnt 0x2" ::: "memory");
        } else {
            asm volatile("s_wait_asynccnt 0x0" ::: "memory");
        }

        const float* b = wbuf + pb * EMBED + lane * 8;
        v4f a0 = *(const v4f*)(b);
        v4f a1 = *(const v4f*)(b + 4);

        float e0 = __expf(a0.x), e1 = __expf(a0.y);
        float e2 = __expf(a0.z), e3 = __expf(a0.w);
        float f0 = __expf(a1.x), f1 = __expf(a1.y);
        float f2 = __expf(a1.z), f3 = __expf(a1.w);

        float s = ((e0 + e1) + (e2 + e3)) + ((f0 + f1) + (f2 + f3));
        // wave32 xor-butterfly: ds_swizzle SWAPX16/8/4/2/1
        XOR_ADD(s, 0x7C10);
        XOR_ADD(s, 0x7C08);
        XOR_ADD(s, 0x7C04);
        XOR_ADD(s, 0x7C02);
        XOR_ADD(s, 0x7C01);
        const float inv = __builtin_amdgcn_rcpf(s);

        const size_t ob = (size_t)r * EMBED + (size_t)lane * 8;
        __builtin_nontemporal_store(a0, (v4f*)(logits + ob));
        __builtin_nontemporal_store(a1, (v4f*)(logits + ob + 4));
        v4f p0 = { e0 * inv, e1 * inv, e2 * inv, e3 * inv };
        v4f p1 = { f0 * inv, f1 * inv, f2 * inv, f3 * inv };
        __builtin_nontemporal_store(p0, (v4f*)(probs + ob));
        __builtin_nontemporal_store(p1, (v4f*)(probs + ob + 4));

        pb ^= 1;
    }
}

extern "C" void kernel_launch(void* const* d_in, const int* in_sizes, int n_in,
                              void* d_out, int out_size, void* d_ws, size_t ws_size,
                              hipStream_t stream) {
    const int*   idx = (const int*)d_in[0];     // [B, T] int32
    const float* emb = (const float*)d_in[1];   // [VOCAB, 256] f32
    float* logits = (float*)d_out;              // [R, 256]
    const int R = in_sizes[0];                  // B*T rows
    float* probs = logits + (size_t)R * EMBED;  // second output, concatenated

    embed_softmax_kernel<<<dim3(BLOCKS), dim3(256), 0, stream>>>(
        idx, emb, logits, probs, R);
}